// BoundaryAwareContrastiveLoss_39195871543751
// MI455X (gfx1250) — compile-verified
//
#include <hip/hip_runtime.h>

// BoundaryAwareContrastiveLoss on MI455X (gfx1250, wave32).
// features: [4, 4096, 64] f32   boundary_map: [4, 4096] f32   out: scalar f32
//
// Fused single pass: per 16x16 Gram tile, K=64 -> 16 x V_WMMA_F32_16X16X4_F32
// split into two independent accumulator chains; B fragments double-buffered
// in registers across the n-tile loop so loads clause-issue ahead of the WMMA
// chain. Distance/hinge epilogue fused, wave-reduced, one atomicAdd per wave.

typedef __attribute__((ext_vector_type(2))) float v2f;
typedef __attribute__((ext_vector_type(8))) float v8f;

#define BATCH        4
#define N_PTS        4096
#define DIMS         64
#define TILES        (N_PTS / 16)            // 256
#define NT_PER_WAVE  16
#define NGROUPS      (TILES / NT_PER_WAVE)   // 16
#define WAVES_PER_BLOCK 8                    // 256 threads = 8 wave32

__global__ void zero_out_kernel(float* __restrict__ out) {
    if (threadIdx.x == 0 && blockIdx.x == 0) out[0] = 0.0f;
}

// One thread per row: sq[row] = sum_d X[row,d]^2   (B*N = 16384 rows)
__global__ void row_sq_kernel(const float* __restrict__ feat,
                              float* __restrict__ sq) {
    const int row = blockIdx.x * blockDim.x + threadIdx.x;
    const float4* p = (const float4*)(feat + (size_t)row * DIMS);
    float s = 0.0f;
#pragma unroll
    for (int i = 0; i < DIMS / 4; ++i) {
        float4 v = p[i];
        s += v.x * v.x + v.y * v.y + v.z * v.z + v.w * v.w;
    }
    sq[row] = s;
}

__global__ void __launch_bounds__(256)
gram_loss_kernel(const float* __restrict__ feat,
                 const float* __restrict__ bmap,
                 const float* __restrict__ sq,
                 float* __restrict__ out) {
    const int lane = threadIdx.x & 31;
    const int w    = blockIdx.x * WAVES_PER_BLOCK + (threadIdx.x >> 5);

    const int b   = w / (TILES * NGROUPS);
    const int rem = w % (TILES * NGROUPS);
    const int mt  = rem / NGROUPS;          // m-tile (0..255)
    const int ng  = rem % NGROUPS;          // n-tile group (0..15)

    const int half = lane >> 4;             // f32 WMMA A/B layout lane halves
    const int r    = lane & 15;

    const float* fb = feat + (size_t)b * N_PTS * DIMS;
    const float* sb = sq   + b * N_PTS;
    const float* mb = bmap + b * N_PTS;

    // A fragments resident for all 16 K-chunks:
    // lane (half,r) holds row mt*16+r, VGPR pair j of chunk kc = K=4kc+2half+j.
    const float* arow = fb + (size_t)(mt * 16 + r) * DIMS + 2 * half;
    v2f afrag[16];
#pragma unroll
    for (int kc = 0; kc < 16; ++kc)
        afrag[kc] = *(const v2f*)(arow + 4 * kc);

    // Epilogue constants: this lane's 8 C elements are (M = i+8*half, N = r).
    float sqm[8], bmm[8];
#pragma unroll
    for (int i = 0; i < 8; ++i) {
        const int m = mt * 16 + i + 8 * half;
        sqm[i] = sb[m];
        bmm[i] = mb[m];
    }

    const int nbase = ng * NT_PER_WAVE;

    // Register double-buffer for B fragments (B = A^T: same load pattern).
    v2f bbuf[2][16];
    {
        const float* brow = fb + (size_t)(nbase * 16 + r) * DIMS + 2 * half;
#pragma unroll
        for (int kc = 0; kc < 16; ++kc)
            bbuf[0][kc] = *(const v2f*)(brow + 4 * kc);
    }

    float acc = 0.0f;
#pragma unroll
    for (int t = 0; t < NT_PER_WAVE; ++t) {
        // Prefetch next tile's 16 B fragments while this tile's WMMAs run.
        if (t + 1 < NT_PER_WAVE) {
            const float* brow =
                fb + (size_t)((nbase + t + 1) * 16 + r) * DIMS + 2 * half;
#pragma unroll
            for (int kc = 0; kc < 16; ++kc)
                bbuf[(t + 1) & 1][kc] = *(const v2f*)(brow + 4 * kc);
        }
        const float sqn = sb[(nbase + t) * 16 + r];
        const float bmn = mb[(nbase + t) * 16 + r];

        // Two independent accumulator chains (even/odd K-chunks) so the
        // matrix pipe is never serialized on a single C RAW chain.
        v8f c0 = {}, c1 = {};
#pragma unroll
        for (int kc = 0; kc < 16; kc += 2) {
            c0 = __builtin_amdgcn_wmma_f32_16x16x4_f32(
                     false, afrag[kc],     false, bbuf[t & 1][kc],
                     (short)0, c0, false, false);
            c1 = __builtin_amdgcn_wmma_f32_16x16x4_f32(
                     false, afrag[kc + 1], false, bbuf[t & 1][kc + 1],
                     (short)0, c1, false, false);
        }

#pragma unroll
        for (int i = 0; i < 8; ++i) {
            float g    = c0[i] + c1[i];
            float d2   = sqm[i] + sqn - 2.0f * g;
            float dist = d2 > 0.0f ? __builtin_sqrtf(d2) : 0.0f;  // safe sqrt
            float pos  = bmm[i] * bmn;
            float hin  = 1.0f - dist;
            hin        = hin > 0.0f ? hin : 0.0f;                 // relu(1-d)
            acc += pos * dist + (1.0f - pos) * hin;
        }
    }

    // wave32 tree reduction
#pragma unroll
    for (int off = 16; off > 0; off >>= 1)
        acc += __shfl_xor(acc, off, 32);

    if (lane == 0) {
        const float scale =
            1.0f / ((float)BATCH * (float)N_PTS * (float)N_PTS);
        atomicAdd(out, acc * scale);
    }
}

extern "C" void kernel_launch(void* const* d_in, const int* in_sizes, int n_in,
                              void* d_out, int out_size, void* d_ws,
                              size_t ws_size, hipStream_t stream) {
    const float* feat = (const float*)d_in[0];   // [4,4096,64]
    const float* bmap = (const float*)d_in[1];   // [4,4096]
    float* out = (float*)d_out;                  // scalar
    float* sq  = (float*)d_ws;                   // 16384 floats (64 KB)

    zero_out_kernel<<<1, 64, 0, stream>>>(out);
    row_sq_kernel<<<(BATCH * N_PTS) / 256, 256, 0, stream>>>(feat, sq);

    const int totalWaves = BATCH * TILES * NGROUPS;      // 16384
    const int blocks = totalWaves / WAVES_PER_BLOCK;     // 2048
    gram_loss_kernel<<<blocks, 256, 0, stream>>>(feat, bmap, sq, out);
}